// SmallestAdditionTransformer_40046275068140
// MI455X (gfx1250) — compile-verified
//
#include <hip/hip_runtime.h>
#include <math.h>

#define SEQ_T 34
#define T_PAD 48
#define VOCAB 14
#define SPB 4                 // sequences per block
#define NTHR (SPB * 64)       // 256 threads = 8 waves

typedef float v2f __attribute__((ext_vector_type(2)));
typedef float v8f __attribute__((ext_vector_type(8)));

__device__ __forceinline__ void ln6(const float* x, const float* g, const float* b, float* o) {
    float mu = 0.f;
#pragma unroll
    for (int d = 0; d < 6; ++d) mu += x[d];
    mu *= (1.f / 6.f);
    float var = 0.f;
#pragma unroll
    for (int d = 0; d < 6; ++d) { float df = x[d] - mu; var += df * df; }
    var *= (1.f / 6.f);
    const float inv = rsqrtf(var + 1e-5f);
#pragma unroll
    for (int d = 0; d < 6; ++d) o[d] = (x[d] - mu) * inv * g[d] + b[d];
}

__global__ __launch_bounds__(NTHR) void tiny_tf_kernel(
    const int* __restrict__ idx,
    const float* __restrict__ tok_emb, const float* __restrict__ pos_enc,
    const float* __restrict__ q_w, const float* __restrict__ k_w, const float* __restrict__ v_w,
    const float* __restrict__ out_w,
    const float* __restrict__ ln1_g, const float* __restrict__ ln1_b,
    const float* __restrict__ ln2_g, const float* __restrict__ ln2_b,
    const float* __restrict__ lnf_g, const float* __restrict__ lnf_b,
    const float* __restrict__ ffn_w1, const float* __restrict__ ffn_b1,
    const float* __restrict__ ffn_w2, const float* __restrict__ ffn_b2,
    const float* __restrict__ head_w,
    float* __restrict__ out, const int Btot)
{
    __shared__ float s_te[VOCAB * 3];
    __shared__ float s_pe[SEQ_T * 3];
    __shared__ float s_qw[18], s_kw[18], s_vw[18], s_ow[36];
    __shared__ float s_l1g[6], s_l1b[6], s_l2g[6], s_l2b[6], s_lfg[6], s_lfb[6];
    __shared__ float s_w1[18], s_b1[3], s_w2[18], s_b2[6], s_hw[18];
    __shared__ float s_Wc[VOCAB * 6];              // folded tok_emb @ head_w  (14x6)
    __shared__ float s_q[SPB][2][T_PAD][4];        // [seq][head][pos][dim], dim3 = 0 pad
    __shared__ float s_k[SPB][2][T_PAD][4];
    __shared__ float s_v[SPB][2][T_PAD][4];
    __shared__ float s_x[SPB][SEQ_T][6];           // residual stream
    __shared__ float s_att[SPB][SEQ_T][6];         // attention output (pre out_w)
    __shared__ float s_logits[SPB * SEQ_T * VOCAB];

    const int tid = threadIdx.x;
    const int blk = blockIdx.x;

    // ---- load all weights into LDS once per block ----
#define CPY(dst, src, n) for (int i = tid; i < (n); i += NTHR) dst[i] = (src)[i]
    CPY(s_te, tok_emb, VOCAB * 3);
    CPY(s_pe, pos_enc, SEQ_T * 3);
    CPY(s_qw, q_w, 18); CPY(s_kw, k_w, 18); CPY(s_vw, v_w, 18);
    CPY(s_ow, out_w, 36);
    CPY(s_l1g, ln1_g, 6); CPY(s_l1b, ln1_b, 6);
    CPY(s_l2g, ln2_g, 6); CPY(s_l2b, ln2_b, 6);
    CPY(s_lfg, lnf_g, 6); CPY(s_lfb, lnf_b, 6);
    CPY(s_w1, ffn_w1, 18); CPY(s_b1, ffn_b1, 3);
    CPY(s_w2, ffn_w2, 18); CPY(s_b2, ffn_b2, 6);
    CPY(s_hw, head_w, 18);
#undef CPY
    __syncthreads();

    // ---- phase 1: embed + ln1 + q/k/v per position (one position per thread) ----
    {
        const int s = tid >> 6;           // local sequence
        const int t = tid & 63;           // position
        const int seq = blk * SPB + s;
        if (t < SEQ_T && seq < Btot) {
            float x[6], h[6];
            const int tok = idx[(size_t)seq * SEQ_T + t];
            x[0] = s_te[tok * 3 + 0]; x[1] = s_te[tok * 3 + 1]; x[2] = s_te[tok * 3 + 2];
            x[3] = s_pe[t * 3 + 0];   x[4] = s_pe[t * 3 + 1];   x[5] = s_pe[t * 3 + 2];
#pragma unroll
            for (int d = 0; d < 6; ++d) s_x[s][t][d] = x[d];
            ln6(x, s_l1g, s_l1b, h);
#pragma unroll
            for (int hh = 0; hh < 2; ++hh) {
#pragma unroll
                for (int d = 0; d < 3; ++d) {
                    const int row = hh * 3 + d;
                    float qv = 0.f, kv = 0.f, vv = 0.f;
#pragma unroll
                    for (int j = 0; j < 3; ++j) {
                        qv += s_qw[row * 3 + j] * h[3 + j];
                        kv += s_kw[row * 3 + j] * h[3 + j];
                        vv += s_vw[row * 3 + j] * h[j];
                    }
                    s_q[s][hh][t][d] = qv; s_k[s][hh][t][d] = kv; s_v[s][hh][t][d] = vv;
                }
                s_q[s][hh][t][3] = 0.f; s_k[s][hh][t][3] = 0.f; s_v[s][hh][t][3] = 0.f;
            }
        } else if (t < T_PAD) {  // zero-pad (rows 34..47, or whole seq if out of range)
#pragma unroll
            for (int hh = 0; hh < 2; ++hh)
#pragma unroll
                for (int d = 0; d < 4; ++d) {
                    s_q[s][hh][t][d] = 0.f; s_k[s][hh][t][d] = 0.f; s_v[s][hh][t][d] = 0.f;
                }
        }
    }
    // fold W = tok_emb(14x3) @ head_w(3x6) -> 14x6
    for (int i = tid; i < VOCAB * 6; i += NTHR) {
        const int c = i / 6, d = i % 6;
        float acc = 0.f;
#pragma unroll
        for (int j = 0; j < 3; ++j) acc += s_te[c * 3 + j] * s_hw[j * 6 + d];
        s_Wc[i] = acc;
    }
    __syncthreads();

    // ---- phase 2: attention via WMMA, one wave per (seq, head) ----
    // S^T = K(16x4) x Q^T(4x16): C-layout puts query q = lane%16,
    // key k = vgpr + 8*(lane>=16); softmax = in-lane reduce + one xor-16 shuffle.
    {
        const int wave = tid >> 5;        // 0..7
        const int s    = wave >> 1;       // local sequence
        const int head = wave & 1;
        const int lane = tid & 31;
        const int n16 = lane & 15;
        const int hi = lane >> 4;         // which half of the K(=4) dim this lane feeds
        const int dbase = hi * 2;         // A/B operand: K-slots {0,1} or {2,3}
        const int kb = hi * 8;            // score rows this lane holds
        const float scale = 0.5773502691896258f;  // 1/sqrt(HEAD_DIM)
#pragma unroll
        for (int qt = 0; qt < 3; ++qt) {
            const int q = qt * 16 + n16;
            v2f bm;                        // B operand = Q^T tile (N = query col)
            bm.x = s_q[s][head][q][dbase];
            bm.y = s_q[s][head][q][dbase + 1];
            float p[24];
#pragma unroll
            for (int kt = 0; kt <= qt; ++kt) {
                const int kpos = kt * 16 + n16;
                v2f am;                    // A operand = K tile (M = key row)
                am.x = s_k[s][head][kpos][dbase];
                am.y = s_k[s][head][kpos][dbase + 1];
                v8f c = {0.f, 0.f, 0.f, 0.f, 0.f, 0.f, 0.f, 0.f};
                c = __builtin_amdgcn_wmma_f32_16x16x4_f32(
                        false, am, false, bm, (short)0, c, false, false);
#pragma unroll
                for (int r = 0; r < 8; ++r) {
                    const int k = kt * 16 + kb + r;
                    p[kt * 8 + r] = (k <= q && k < SEQ_T && q < SEQ_T)
                                        ? c[r] * scale : -1e30f;
                }
            }
            float m = -1e30f;
#pragma unroll
            for (int kt = 0; kt <= qt; ++kt)
#pragma unroll
                for (int r = 0; r < 8; ++r) m = fmaxf(m, p[kt * 8 + r]);
            m = fmaxf(m, __shfl_xor(m, 16, 32));
            float sum = 0.f, a0 = 0.f, a1 = 0.f, a2 = 0.f;
#pragma unroll
            for (int kt = 0; kt <= qt; ++kt)
#pragma unroll
                for (int r = 0; r < 8; ++r) {
                    const int k = kt * 16 + kb + r;
                    const float e = __expf(p[kt * 8 + r] - m);
                    sum += e;
                    a0 += e * s_v[s][head][k][0];
                    a1 += e * s_v[s][head][k][1];
                    a2 += e * s_v[s][head][k][2];
                }
            sum += __shfl_xor(sum, 16, 32);
            a0  += __shfl_xor(a0, 16, 32);
            a1  += __shfl_xor(a1, 16, 32);
            a2  += __shfl_xor(a2, 16, 32);
            if (hi == 0 && q < SEQ_T) {
                const float inv = 1.f / sum;
                s_att[s][q][head * 3 + 0] = a0 * inv;
                s_att[s][q][head * 3 + 1] = a1 * inv;
                s_att[s][q][head * 3 + 2] = a2 * inv;
            }
        }
    }
    __syncthreads();

    // ---- phase 3: out-proj residual, FFN (exact GELU), final LN, logits ----
    {
        const int s = tid >> 6;
        const int t = tid & 63;
        const int seq = blk * SPB + s;
        if (t < SEQ_T && seq < Btot) {
            float x[6], h2[6], xf[6];
#pragma unroll
            for (int d = 0; d < 6; ++d) x[d] = s_x[s][t][d];
#pragma unroll
            for (int i = 0; i < 6; ++i) {
                float acc = 0.f;
#pragma unroll
                for (int j = 0; j < 6; ++j) acc += s_att[s][t][j] * s_ow[i * 6 + j];
                x[i] += acc;
            }
            ln6(x, s_l2g, s_l2b, h2);
            float g[3];
#pragma unroll
            for (int kk = 0; kk < 3; ++kk) {
                float pre = s_b1[kk];
#pragma unroll
                for (int j = 0; j < 6; ++j) pre += s_w1[kk * 6 + j] * h2[j];
                g[kk] = 0.5f * pre * (1.f + erff(pre * 0.7071067811865476f));
            }
#pragma unroll
            for (int i = 0; i < 6; ++i) {
                float acc = s_b2[i];
#pragma unroll
                for (int kk = 0; kk < 3; ++kk) acc += s_w2[i * 3 + kk] * g[kk];
                x[i] += acc;
            }
            ln6(x, s_lfg, s_lfb, xf);
#pragma unroll
            for (int c = 0; c < VOCAB; ++c) {
                float acc = 0.f;
#pragma unroll
                for (int d = 0; d < 6; ++d) acc += xf[d] * s_Wc[c * 6 + d];
                s_logits[(s * SEQ_T + t) * VOCAB + c] = acc;
            }
        }
    }
    __syncthreads();

    // ---- coalesced store: 1904 contiguous floats per block ----
    const size_t base = (size_t)blk * (SPB * SEQ_T * VOCAB);
    const size_t total = (size_t)Btot * (SEQ_T * VOCAB);
    for (int i = tid; i < SPB * SEQ_T * VOCAB; i += NTHR) {
        const size_t gi = base + i;
        if (gi < total) out[gi] = s_logits[i];
    }
}

extern "C" void kernel_launch(void* const* d_in, const int* in_sizes, int n_in,
                              void* d_out, int out_size, void* d_ws, size_t ws_size,
                              hipStream_t stream) {
    (void)n_in; (void)out_size; (void)d_ws; (void)ws_size;
    const int B = in_sizes[0] / SEQ_T;
    const int nblk = (B + SPB - 1) / SPB;
    tiny_tf_kernel<<<nblk, NTHR, 0, stream>>>(
        (const int*)d_in[0],
        (const float*)d_in[1], (const float*)d_in[2],
        (const float*)d_in[3], (const float*)d_in[4], (const float*)d_in[5],
        (const float*)d_in[6],
        (const float*)d_in[7], (const float*)d_in[8],
        (const float*)d_in[9], (const float*)d_in[10],
        (const float*)d_in[11], (const float*)d_in[12],
        (const float*)d_in[13], (const float*)d_in[14],
        (const float*)d_in[15], (const float*)d_in[16],
        (const float*)d_in[17],
        (float*)d_out, B);
}